// GINNet_78211354460181
// MI455X (gfx1250) — compile-verified
//
#include <hip/hip_runtime.h>

#define NN   100000
#define NE   600000
#define NG   5000
#define IND  11
#define HID  128
#define K0P  16   // layer-0 K padded 11 -> 16

typedef float v2f __attribute__((ext_vector_type(2)));
typedef float v8f __attribute__((ext_vector_type(8)));

// ---------------------------------------------------------------- utilities
__global__ void zero_kernel(float* __restrict__ p, int n) {
    int i = blockIdx.x * blockDim.x + threadIdx.x;
    if (i < n) p[i] = 0.f;
}

// h1 = (1+eps)*x, zero-padded to Dpad columns (eps read from device scalar)
__global__ void prep_kernel(const float* __restrict__ xin, const float* __restrict__ eps_p,
                            float* __restrict__ h1, int D, int Dpad, int nN) {
    int i = blockIdx.x * blockDim.x + threadIdx.x;
    if (i >= nN * Dpad) return;
    int n = i / Dpad, d = i - n * Dpad;
    float v = 0.f;
    if (d < D) v = (1.f + eps_p[0]) * xin[(size_t)n * D + d];
    h1[i] = v;
}

// msg = relu(x[src] + edge_attr@ew + eb) ; scatter-add into aggr[dst]
__global__ void edge_kernel(const float* __restrict__ xin, int xs,
                            const float* __restrict__ ea,
                            const float* __restrict__ ew, const float* __restrict__ eb,
                            const int* __restrict__ src, const int* __restrict__ dst,
                            float* __restrict__ aggr, int as, int D) {
    int i = blockIdx.x * blockDim.x + threadIdx.x;
    if (i >= NE * D) return;
    int e = i / D, d = i - e * D;
    float a0 = ea[e * 3 + 0], a1 = ea[e * 3 + 1], a2 = ea[e * 3 + 2];
    float lin = eb[d] + a0 * ew[d] + a1 * ew[D + d] + a2 * ew[2 * D + d];
    float msg = xin[(size_t)src[e] * xs + d] + lin;
    msg = fmaxf(msg, 0.f);
    atomicAdd(&aggr[(size_t)dst[e] * as + d], msg);
}

// ------------------------------------------------------------- WMMA GEMM
// out[n_rows x 128] = act(A[n_rows x K] @ W[K x 128] + bias), act = relu if relu_out.
// W rows >= realKW treated as zero (layer-0 padding). 256 threads = 8 waves; each
// wave computes a 16-row x 128-col tile via V_WMMA_F32_16X16X4_F32.
// LDS holds W pair-interleaved: lds_w[p*128 + n] = {W[2p][n], W[2p+1][n]}, so each
// B operand is one contiguous ds_load_b64 into an even-aligned VGPR pair.
__global__ void __launch_bounds__(256)
gemm_kernel(const float* __restrict__ A, const float* __restrict__ W,
            const float* __restrict__ bias, float* __restrict__ out,
            int n_rows, int K, int realKW, int relu_out) {
    __shared__ v2f lds_w[(HID / 2) * HID];  // up to 64x128 pairs = 64 KB
    int tid = threadIdx.x;
    int pairs = K >> 1;
    for (int idx = tid; idx < pairs * HID; idx += blockDim.x) {
        int p = idx / HID, n = idx - p * HID;
        v2f w;
        w.x = (2 * p     < realKW) ? W[(2 * p)     * HID + n] : 0.f;
        w.y = (2 * p + 1 < realKW) ? W[(2 * p + 1) * HID + n] : 0.f;
        lds_w[idx] = w;
    }
    __syncthreads();

    int lane = tid & 31;
    int wv   = tid >> 5;
    int m    = lane & 15;     // row-in-tile for A, col-in-tile for B/C/D
    int kh   = lane >> 4;     // K-pair selector (0 -> K{0,1}, 1 -> K{2,3})
    int row0 = blockIdx.x * 128 + wv * 16;
    int row  = row0 + m;
    // Clamp instead of predicating: an A row only feeds its own output row, and
    // out-of-range output rows are never stored, so reading a valid stand-in row
    // is safe and keeps the inner loop branch-free.
    int rrow = row < n_rows ? row : n_rows - 1;
    const v2f* __restrict__ arow2 = (const v2f*)(A + (size_t)rrow * K);

    v8f acc[8] = {};

    for (int kb = 0; kb < K; kb += 4) {
        int p = (kb >> 1) + kh;          // pair index for this lane's K{.,.+1}
        v2f a = arow2[p];                // one global_load_b64
#pragma unroll
        for (int t = 0; t < 8; ++t) {
            v2f b = lds_w[p * HID + t * 16 + m];   // one ds_load_b64
            acc[t] = __builtin_amdgcn_wmma_f32_16x16x4_f32(
                false, a, false, b, (short)0, acc[t], false, false);
        }
    }

#pragma unroll
    for (int t = 0; t < 8; ++t) {
        int col  = t * 16 + m;
        float bv = bias[col];
#pragma unroll
        for (int r = 0; r < 8; ++r) {
            int orow = row0 + 8 * kh + r;   // D layout: M = r + 8*(lane/16)
            if (orow < n_rows) {
                float v = acc[t][r] + bv;
                if (relu_out) v = fmaxf(v, 0.f);
                out[(size_t)orow * HID + col] = v;
            }
        }
    }
}

// -------------------------------------------------------------- BatchNorm
__global__ void bn_stats_kernel(const float* __restrict__ h, int nN,
                                float* __restrict__ csum, float* __restrict__ csq) {
    int col = threadIdx.x;                 // 128 threads, coalesced across cols
    int r0 = blockIdx.x * 1024;
    int r1 = r0 + 1024; if (r1 > nN) r1 = nN;
    float s = 0.f, q = 0.f;
    for (int r = r0; r < r1; ++r) {
        float v = h[(size_t)r * HID + col];
        s += v; q += v * v;
    }
    atomicAdd(&csum[col], s);
    atomicAdd(&csq[col], q);
}

// h = relu((h-mu)*rsqrt(var+eps)*gamma + beta) (+ residual), in place
__global__ void bn_apply_kernel(float* __restrict__ h, const float* __restrict__ csum,
                                const float* __restrict__ csq,
                                const float* __restrict__ gamma, const float* __restrict__ beta,
                                const float* __restrict__ hres, int nN) {
    int i = blockIdx.x * blockDim.x + threadIdx.x;
    if (i >= nN * HID) return;
    int d = i & (HID - 1);
    float inv_n = 1.f / (float)nN;
    float mu  = csum[d] * inv_n;
    float var = csq[d] * inv_n - mu * mu;
    float v = (h[i] - mu) * rsqrtf(var + 1e-5f) * gamma[d] + beta[d];
    v = fmaxf(v, 0.f);
    if (hres) v += hres[i];
    h[i] = v;
}

// ------------------------------------------------------------------ pooling
__global__ void pool_kernel(const float* __restrict__ h, const int* __restrict__ batch,
                            float* __restrict__ psum, float* __restrict__ pcnt) {
    int i = blockIdx.x * blockDim.x + threadIdx.x;
    if (i >= NN * HID) return;
    int n = i >> 7, d = i & 127;
    int g = batch[n];
    atomicAdd(&psum[(size_t)g * HID + d], h[i]);
    if (d == 0) atomicAdd(&pcnt[g], 1.f);
}

__global__ void pool_div_kernel(float* __restrict__ psum, const float* __restrict__ pcnt) {
    int i = blockIdx.x * blockDim.x + threadIdx.x;
    if (i >= NG * HID) return;
    int g = i >> 7;
    psum[i] = psum[i] / fmaxf(pcnt[g], 1.f);
}

__global__ void head_out_kernel(const float* __restrict__ z, const float* __restrict__ hw3,
                                const float* __restrict__ hb3, float* __restrict__ out) {
    int g = blockIdx.x * blockDim.x + threadIdx.x;
    if (g >= NG) return;
    float s = hb3[0];
    for (int d = 0; d < HID; ++d) s += z[(size_t)g * HID + d] * hw3[d];
    out[g] = s;
}

// ------------------------------------------------------------------ driver
extern "C" void kernel_launch(void* const* d_in, const int* in_sizes, int n_in,
                              void* d_out, int out_size, void* d_ws, size_t ws_size,
                              hipStream_t stream) {
    const float* x    = (const float*)d_in[0];
    const float* ea   = (const float*)d_in[1];
    const float* ew0  = (const float*)d_in[2];
    const float* eb0  = (const float*)d_in[3];
    const float* w1_0 = (const float*)d_in[4];
    const float* b1_0 = (const float*)d_in[5];
    const float* w2_0 = (const float*)d_in[6];
    const float* b2_0 = (const float*)d_in[7];
    const float* g0   = (const float*)d_in[8];
    const float* be0  = (const float*)d_in[9];
    const float* eps0 = (const float*)d_in[10];
    const float* ewR  = (const float*)d_in[11];
    const float* ebR  = (const float*)d_in[12];
    const float* w1R  = (const float*)d_in[13];
    const float* b1R  = (const float*)d_in[14];
    const float* w2R  = (const float*)d_in[15];
    const float* b2R  = (const float*)d_in[16];
    const float* gR   = (const float*)d_in[17];
    const float* beR  = (const float*)d_in[18];
    const float* epsR = (const float*)d_in[19];
    const float* hw1  = (const float*)d_in[20];
    const float* hb1  = (const float*)d_in[21];
    const float* hw2  = (const float*)d_in[22];
    const float* hb2  = (const float*)d_in[23];
    const float* hw3  = (const float*)d_in[24];
    const float* hb3  = (const float*)d_in[25];
    const int*   eidx = (const int*)d_in[26];
    const int*   batch= (const int*)d_in[27];
    const int* src = eidx;
    const int* dst = eidx + NE;

    float* ws = (float*)d_ws;
    const size_t BIG = (size_t)NN * HID;
    float* P0   = ws;                 // node features ping
    float* P1   = P0 + BIG;           // node features pong / aggr / h2
    float* MID  = P1 + BIG;           // MLP hidden
    float* csum = MID + BIG;          // [128]
    float* csq  = csum + HID;         // [128]
    float* psum = csq + HID;          // [NG*128]
    float* pcnt = psum + (size_t)NG * HID;  // [NG]
    float* z1   = pcnt + NG;          // [NG*128]
    float* z2   = z1 + (size_t)NG * HID;    // [NG*128]

    const int T = 256;
    const int gNodeHID = (NN * HID + T - 1) / T;
    const int gGemmN   = (NN + 127) / 128;
    const int gGemmG   = (NG + 127) / 128;

    // ---------------- layer 0 (11 -> 128), no residual ----------------
    prep_kernel<<<(NN * K0P + T - 1) / T, T, 0, stream>>>(x, eps0, P1, IND, K0P, NN);
    edge_kernel<<<(NE * IND + T - 1) / T, T, 0, stream>>>(x, IND, ea, ew0, eb0,
                                                          src, dst, P1, K0P, IND);
    gemm_kernel<<<gGemmN, T, 0, stream>>>(P1, w1_0, b1_0, MID, NN, K0P, IND, 1);
    gemm_kernel<<<gGemmN, T, 0, stream>>>(MID, w2_0, b2_0, P0, NN, HID, HID, 0);
    zero_kernel<<<1, 256, 0, stream>>>(csum, 2 * HID);
    bn_stats_kernel<<<(NN + 1023) / 1024, HID, 0, stream>>>(P0, NN, csum, csq);
    bn_apply_kernel<<<gNodeHID, T, 0, stream>>>(P0, csum, csq, g0, be0, nullptr, NN);

    // ---------------- layers 1..4 (128 -> 128), residual ----------------
    float* cur = P0; float* oth = P1;
    for (int l = 0; l < 4; ++l) {
        prep_kernel<<<gNodeHID, T, 0, stream>>>(cur, epsR + l, oth, HID, HID, NN);
        edge_kernel<<<(NE * HID + T - 1) / T, T, 0, stream>>>(cur, HID, ea,
            ewR + (size_t)l * 3 * HID, ebR + (size_t)l * HID, src, dst, oth, HID, HID);
        gemm_kernel<<<gGemmN, T, 0, stream>>>(oth, w1R + (size_t)l * HID * HID,
                                              b1R + (size_t)l * HID, MID, NN, HID, HID, 1);
        gemm_kernel<<<gGemmN, T, 0, stream>>>(MID, w2R + (size_t)l * HID * HID,
                                              b2R + (size_t)l * HID, oth, NN, HID, HID, 0);
        zero_kernel<<<1, 256, 0, stream>>>(csum, 2 * HID);
        bn_stats_kernel<<<(NN + 1023) / 1024, HID, 0, stream>>>(oth, NN, csum, csq);
        bn_apply_kernel<<<gNodeHID, T, 0, stream>>>(oth, csum, csq,
            gR + (size_t)l * HID, beR + (size_t)l * HID, cur, NN);
        float* tmp = cur; cur = oth; oth = tmp;
    }

    // ---------------- global mean pool + head MLP ----------------
    zero_kernel<<<((NG * HID + NG) + T - 1) / T, T, 0, stream>>>(psum, NG * HID + NG);
    pool_kernel<<<gNodeHID, T, 0, stream>>>(cur, batch, psum, pcnt);
    pool_div_kernel<<<(NG * HID + T - 1) / T, T, 0, stream>>>(psum, pcnt);
    gemm_kernel<<<gGemmG, T, 0, stream>>>(psum, hw1, hb1, z1, NG, HID, HID, 1);
    gemm_kernel<<<gGemmG, T, 0, stream>>>(z1, hw2, hb2, z2, NG, HID, HID, 1);
    head_out_kernel<<<(NG + T - 1) / T, T, 0, stream>>>(z2, hw3, hb3, (float*)d_out);
}